// HierarchicalQuantizedEmbedding_90658169684714
// MI455X (gfx1250) — compile-verified
//
#include <hip/hip_runtime.h>
#include <cstdint>
#include <cstddef>

#define DIM 1024
#define TPB 256

// ---------------------------------------------------------------------------
// Workspace layout: ws[0..2] = abs-max bits (as uint) for tiers 1,2,3.
// ---------------------------------------------------------------------------
__global__ void hqe_init_ws(unsigned* __restrict__ w) {
  if (threadIdx.x < 3) w[threadIdx.x] = 0u;
}

// ---------------------------------------------------------------------------
// Per-tier abs-max reduction over |tw[r,c] * ts[c]|.
// n4 = number of float4 elements (rows * 256). Column float4 index = i & 255.
// Non-negative floats order-preserve under uint bitcast -> atomicMax(uint).
// ---------------------------------------------------------------------------
__global__ __launch_bounds__(TPB) void hqe_absmax(const float4* __restrict__ tw,
                                                  const float4* __restrict__ ts,
                                                  long n4,
                                                  unsigned* __restrict__ out) {
  float m = 0.0f;
  const long stride = (long)gridDim.x * TPB;
  for (long i = (long)blockIdx.x * TPB + threadIdx.x; i < n4; i += stride) {
    float4 w = tw[i];
    float4 s = ts[i & 255];
    m = fmaxf(m, fabsf(w.x * s.x));
    m = fmaxf(m, fabsf(w.y * s.y));
    m = fmaxf(m, fabsf(w.z * s.z));
    m = fmaxf(m, fabsf(w.w * s.w));
  }
  __shared__ float red[TPB];
  red[threadIdx.x] = m;
  __syncthreads();
  for (int off = TPB / 2; off > 0; off >>= 1) {
    if (threadIdx.x < off)
      red[threadIdx.x] = fmaxf(red[threadIdx.x], red[threadIdx.x + off]);
    __syncthreads();
  }
  if (threadIdx.x == 0) atomicMax(out, __float_as_uint(red[0]));
}

// ---------------------------------------------------------------------------
// CDNA5 async global -> LDS staging (16B per lane). VADDR = 64-bit global
// address (VGPR pair), dst operand = LDS byte address (low 32 bits of the
// flat shared pointer). Tracked by ASYNCcnt.
// ---------------------------------------------------------------------------
__device__ __forceinline__ void hqe_stage16(const float* gsrc, const float* lds_flat) {
  unsigned lds = (unsigned)(uintptr_t)lds_flat;                 // LDS byte offset
  unsigned long long ga = (unsigned long long)(uintptr_t)gsrc;  // global address
  asm volatile("global_load_async_to_lds_b128 %0, %1, off"
               :
               : "v"(lds), "v"(ga)
               : "memory");
}

// Fake-quant: clip(rint(x/scale), -mv, mv) * scale  (rint = round-half-even,
// matching jnp.round).
__device__ __forceinline__ float hqe_fq(float x, float sc, float mv) {
  float q = rintf(x / sc);
  q = fminf(fmaxf(q, -mv), mv);
  return q * sc;
}

struct RowInfo {
  const float* src;
  const float4* ts;
  float sc;
  float mv;
  int tier;
};

// ---------------------------------------------------------------------------
// Gather with on-the-fly quantization. One 4KB row per block iteration,
// double-buffered in LDS via async DMA: stage row(i+1) while quantizing
// row(i). s_wait_asynccnt 1 => current buffer's DMA complete (in-order),
// barrier publishes it to all 8 waves of the block.
// ---------------------------------------------------------------------------
__global__ __launch_bounds__(TPB) void hqe_gather(
    const int* __restrict__ ids,
    const float* __restrict__ tw0, const float* __restrict__ tw1,
    const float* __restrict__ tw2, const float* __restrict__ tw3,
    const float* __restrict__ ts0, const float* __restrict__ ts1,
    const float* __restrict__ ts2, const float* __restrict__ ts3,
    const unsigned* __restrict__ absbits,
    float* __restrict__ out,
    int n_ids, int b0, int b1, int b2) {
  __shared__ __align__(16) float buf[2][DIM];
  const int tid = threadIdx.x;

  // Per-tier quantization scales: max(absmax, 1e-8) / max_val
  const float s1 = fmaxf(__uint_as_float(absbits[0]), 1e-8f) / 127.0f;
  const float s2 = fmaxf(__uint_as_float(absbits[1]), 1e-8f) / 31.0f;
  const float s3 = fmaxf(__uint_as_float(absbits[2]), 1e-8f) / 7.0f;

  auto resolve = [&](int row) -> RowInfo {
    RowInfo ri;
    int id = ids[row];
    if (id < b0)      { ri.src = tw0 + (size_t)id * DIM;        ri.ts = (const float4*)ts0; ri.tier = 0; ri.sc = 1.0f; ri.mv = 0.0f;   }
    else if (id < b1) { ri.src = tw1 + (size_t)(id - b0) * DIM; ri.ts = (const float4*)ts1; ri.tier = 1; ri.sc = s1;   ri.mv = 127.0f; }
    else if (id < b2) { ri.src = tw2 + (size_t)(id - b1) * DIM; ri.ts = (const float4*)ts2; ri.tier = 2; ri.sc = s2;   ri.mv = 31.0f;  }
    else              { ri.src = tw3 + (size_t)(id - b2) * DIM; ri.ts = (const float4*)ts3; ri.tier = 3; ri.sc = s3;   ri.mv = 7.0f;   }
    return ri;
  };

  int row = blockIdx.x;
  if (row >= n_ids) return;
  const int rstride = gridDim.x;

  RowInfo cur = resolve(row);
  hqe_stage16(cur.src + tid * 4, &buf[0][tid * 4]);

  int parity = 0;
  while (true) {
    const int next = row + rstride;
    const bool hasNext = next < n_ids;
    RowInfo nxt;
    if (hasNext) {
      nxt = resolve(next);
      hqe_stage16(nxt.src + tid * 4, &buf[parity ^ 1][tid * 4]);
      asm volatile("s_wait_asynccnt 1" ::: "memory");  // current row landed
    } else {
      asm volatile("s_wait_asynccnt 0" ::: "memory");  // drain
    }
    __syncthreads();  // publish DMA'd row across all waves

    float4 v = *(const float4*)&buf[parity][tid * 4];
    float4 s = cur.ts[tid];
    float x0 = v.x * s.x, x1 = v.y * s.y, x2 = v.z * s.z, x3 = v.w * s.w;
    float4 o;
    if (cur.tier == 0) {
      o.x = x0; o.y = x1; o.z = x2; o.w = x3;
    } else {
      o.x = hqe_fq(x0, cur.sc, cur.mv);
      o.y = hqe_fq(x1, cur.sc, cur.mv);
      o.z = hqe_fq(x2, cur.sc, cur.mv);
      o.w = hqe_fq(x3, cur.sc, cur.mv);
    }
    ((float4*)(out + (size_t)row * DIM))[tid] = o;

    __syncthreads();  // buffer reusable before next overwrite
    if (!hasNext) break;
    row = next;
    cur = nxt;
    parity ^= 1;
  }
}

// ---------------------------------------------------------------------------
// Host launcher. Input order: input_ids, tw0..tw3, ts0..ts3.
// ---------------------------------------------------------------------------
extern "C" void kernel_launch(void* const* d_in, const int* in_sizes, int n_in,
                              void* d_out, int out_size, void* d_ws, size_t ws_size,
                              hipStream_t stream) {
  const int*   ids = (const int*)d_in[0];
  const float* tw0 = (const float*)d_in[1];
  const float* tw1 = (const float*)d_in[2];
  const float* tw2 = (const float*)d_in[3];
  const float* tw3 = (const float*)d_in[4];
  const float* ts0 = (const float*)d_in[5];
  const float* ts1 = (const float*)d_in[6];
  const float* ts2 = (const float*)d_in[7];
  const float* ts3 = (const float*)d_in[8];
  unsigned* absbits = (unsigned*)d_ws;
  float* out = (float*)d_out;

  const int n_ids = in_sizes[0];
  const int r0 = in_sizes[1] / DIM;
  const int r1 = in_sizes[2] / DIM;
  const int r2 = in_sizes[3] / DIM;
  const int bnd0 = r0;
  const int bnd1 = r0 + r1;
  const int bnd2 = r0 + r1 + r2;

  hqe_init_ws<<<1, 32, 0, stream>>>(absbits);

  auto launch_absmax = [&](const float* tw, const float* ts, int nelem, unsigned* dst) {
    long n4 = (long)nelem / 4;
    int blocks = (int)((n4 + TPB - 1) / TPB);
    if (blocks > 2048) blocks = 2048;
    if (blocks < 1) blocks = 1;
    hqe_absmax<<<blocks, TPB, 0, stream>>>((const float4*)tw, (const float4*)ts, n4, dst);
  };
  launch_absmax(tw1, ts1, in_sizes[2], absbits + 0);  // 8-bit tier
  launch_absmax(tw2, ts2, in_sizes[3], absbits + 1);  // 6-bit tier
  launch_absmax(tw3, ts3, in_sizes[4], absbits + 2);  // 4-bit tier

  int gblocks = n_ids < 2048 ? n_ids : 2048;
  if (gblocks < 1) gblocks = 1;
  hqe_gather<<<gblocks, TPB, 0, stream>>>(ids, tw0, tw1, tw2, tw3,
                                          ts0, ts1, ts2, ts3,
                                          absbits, out, n_ids, bnd0, bnd1, bnd2);
}